// find_NN_51230369907083
// MI455X (gfx1250) — compile-verified
//
#include <hip/hip_runtime.h>
#include <stdint.h>

#define NB      8192      // histogram buckets = bits(d2) >> 19  (sign=0, exp8, mant4)
#define BSHIFT  19
#define CAP     8192      // candidate buffer cap (pathological safety)
#define BLK     256
#define GRID1   2048
#define KSEL    64

// ---- identical d2 in both streaming passes (pinned ops, no contraction drift) ----
__device__ __forceinline__ float d2_of(float x, float y, float z,
                                       float px, float py, float pz) {
    float dx = x - px, dy = y - py, dz = z - pz;
    return __fmaf_rn(dx, dx, __fmaf_rn(dy, dy, __fmul_rn(dz, dz)));
}

__device__ __forceinline__ uint32_t lds_off_of(const void* p) {
    // low 32 bits of a flat shared pointer = LDS byte address
    return (uint32_t)(uintptr_t)p;
}

// Issue 3 x 16B async global->LDS copies per lane (48B = 4 points).
// INST_OFFSET applies to BOTH lds and global addresses (ISA 08 §4.4).
__device__ __forceinline__ void issue3(uint32_t lds, const char* g) {
    asm volatile(
        "global_load_async_to_lds_b128 %0, %1, off\n\t"
        "global_load_async_to_lds_b128 %0, %1, off offset:16\n\t"
        "global_load_async_to_lds_b128 %0, %1, off offset:32"
        :: "v"(lds), "v"(g) : "memory");
}

__global__ void knn_zero_kernel(uint32_t* ws, int n) {
    int i = blockIdx.x * blockDim.x + threadIdx.x;
    if (i < n) ws[i] = 0u;
}

// ---- pass 1: streamed histogram via double-buffered async-to-LDS ----
__global__ __launch_bounds__(BLK)
void knn_hist_kernel(const float* __restrict__ pts, const float* __restrict__ P1,
                     uint32_t* __restrict__ ghist, int N) {
    __shared__ uint32_t lh[NB];                 // 32 KB
    __shared__ float4   stage[2][BLK * 3];      // 24 KB (48B/thread/buffer)

    for (int i = threadIdx.x; i < NB; i += BLK) lh[i] = 0u;
    __syncthreads();

    const float px = P1[0], py = P1[1], pz = P1[2];
    const int   tilePts = BLK * 4;              // 1024 points / block-tile
    const int   tiles   = N / tilePts;
    const char* gbase   = (const char*)pts;

    const uint32_t l0 = lds_off_of(&stage[0][threadIdx.x * 3]);
    const uint32_t l1 = lds_off_of(&stage[1][threadIdx.x * 3]);

    int  tile = blockIdx.x;
    bool have = tile < tiles;
    if (have)
        issue3(l0, gbase + ((size_t)tile * tilePts + threadIdx.x * 4) * 12u);

    int buf = 0;
    while (have) {
        const int  next     = tile + gridDim.x;
        const bool haveNext = next < tiles;
        if (haveNext) {
            // WAR: prior DS reads of the other buffer must land before async writes it
            asm volatile("s_wait_dscnt 0" ::: "memory");
            issue3(buf ? l0 : l1,
                   gbase + ((size_t)next * tilePts + threadIdx.x * 4) * 12u);
            asm volatile("s_wait_asynccnt 3" ::: "memory"); // current buffer's 3 done
        } else {
            asm volatile("s_wait_asynccnt 0" ::: "memory");
        }

        const float4* s = &stage[buf][threadIdx.x * 3];
        float4 a = s[0], b = s[1], c = s[2];
        uint32_t b0 = __float_as_uint(d2_of(a.x, a.y, a.z, px, py, pz)) >> BSHIFT;
        uint32_t b1 = __float_as_uint(d2_of(a.w, b.x, b.y, px, py, pz)) >> BSHIFT;
        uint32_t b2 = __float_as_uint(d2_of(b.z, b.w, c.x, px, py, pz)) >> BSHIFT;
        uint32_t b3 = __float_as_uint(d2_of(c.y, c.z, c.w, px, py, pz)) >> BSHIFT;
        atomicAdd(&lh[b0], 1u);
        atomicAdd(&lh[b1], 1u);
        atomicAdd(&lh[b2], 1u);
        atomicAdd(&lh[b3], 1u);

        tile = next; have = haveNext; buf ^= 1;
    }

    // tail (N % 1024), block 0 only, direct loads
    if (blockIdx.x == 0) {
        for (int i = tiles * tilePts + threadIdx.x; i < N; i += BLK) {
            float x = pts[(size_t)i * 3 + 0];
            float y = pts[(size_t)i * 3 + 1];
            float z = pts[(size_t)i * 3 + 2];
            atomicAdd(&lh[__float_as_uint(d2_of(x, y, z, px, py, pz)) >> BSHIFT], 1u);
        }
    }

    __syncthreads();
    for (int i = threadIdx.x; i < NB; i += BLK) {
        uint32_t v = lh[i];
        if (v) atomicAdd(&ghist[i], v);
    }
}

// ---- pass 2: find threshold bucket T (cumulative count >= K) ----
__global__ __launch_bounds__(256)
void knn_thresh_kernel(const uint32_t* __restrict__ ghist,
                       uint32_t* __restrict__ ctrl, const int* __restrict__ Kp) {
    __shared__ uint32_t coarse[256];
    const uint32_t K = (uint32_t)(*Kp);
    uint32_t s = 0;
    for (int i = 0; i < 32; ++i) s += ghist[threadIdx.x * 32 + i];
    coarse[threadIdx.x] = s;
    __syncthreads();
    if (threadIdx.x == 0) {
        uint32_t cum = 0; int cb = 255;
        for (int i = 0; i < 256; ++i) {
            if (cum + coarse[i] >= K) { cb = i; break; }
            cum += coarse[i];
        }
        uint32_t T = (uint32_t)cb * 32 + 31;
        for (int i = cb * 32; i < cb * 32 + 32; ++i) {
            cum += ghist[i];
            if (cum >= K) { T = (uint32_t)i; break; }
        }
        ctrl[0] = T;
    }
}

// ---- pass 3: collect candidates with bucket <= T (L2-resident re-read) ----
__global__ __launch_bounds__(BLK)
void knn_collect_kernel(const float* __restrict__ pts, const float* __restrict__ P1,
                        const uint32_t* __restrict__ ctrl, uint32_t* __restrict__ cnt,
                        uint64_t* __restrict__ cand, int N) {
    const uint32_t T = ctrl[0];
    const float px = P1[0], py = P1[1], pz = P1[2];
    const int gt = blockIdx.x * BLK + threadIdx.x;
    const int nth = gridDim.x * BLK;
    const float4* p4 = (const float4*)pts;
    const int groups = N / 4;

    for (int g = gt; g < groups; g += nth) {
        float4 a = p4[3 * g], b = p4[3 * g + 1], c = p4[3 * g + 2];
        float d2[4] = { d2_of(a.x, a.y, a.z, px, py, pz),
                        d2_of(a.w, b.x, b.y, px, py, pz),
                        d2_of(b.z, b.w, c.x, px, py, pz),
                        d2_of(c.y, c.z, c.w, px, py, pz) };
        #pragma unroll
        for (int j = 0; j < 4; ++j) {
            uint32_t bits = __float_as_uint(d2[j]);
            if ((bits >> BSHIFT) <= T) {
                uint32_t pos = atomicAdd(cnt, 1u);
                if (pos < CAP)
                    cand[pos] = ((uint64_t)bits << 32) | (uint32_t)(4 * g + j);
            }
        }
    }
    for (int i = groups * 4 + gt; i < N; i += nth) {
        float x = pts[(size_t)i * 3], y = pts[(size_t)i * 3 + 1], z = pts[(size_t)i * 3 + 2];
        uint32_t bits = __float_as_uint(d2_of(x, y, z, px, py, pz));
        if ((bits >> BSHIFT) <= T) {
            uint32_t pos = atomicAdd(cnt, 1u);
            if (pos < CAP) cand[pos] = ((uint64_t)bits << 32) | (uint32_t)i;
        }
    }
}

// ---- pass 4: exact rank of candidates, gather coords, emit output ----
__global__ __launch_bounds__(256)
void knn_final_kernel(const float* __restrict__ pts, const uint32_t* __restrict__ ctrl,
                      const uint64_t* __restrict__ cand, const int* __restrict__ Kp,
                      float* __restrict__ out) {
    __shared__ uint64_t sc[CAP];   // 64 KB (fine inside 320 KB WGP LDS)
    const uint32_t K = (uint32_t)(*Kp);
    uint32_t M = ctrl[1];
    if (M > CAP) M = CAP;

    for (uint32_t i = threadIdx.x; i < 4 * K; i += 256) out[i] = 0.0f; // cover M<K edge
    for (uint32_t i = threadIdx.x; i < M; i += 256) sc[i] = cand[i];
    __threadfence_block();
    __syncthreads();

    for (uint32_t i = threadIdx.x; i < M; i += 256) {
        uint64_t k = sc[i];
        uint32_t r = 0;
        for (uint32_t j = 0; j < M; ++j) r += (sc[j] < k) ? 1u : 0u;
        if (r < K) {
            uint32_t idx = (uint32_t)k;
            out[3 * r + 0] = pts[(size_t)idx * 3 + 0];
            out[3 * r + 1] = pts[(size_t)idx * 3 + 1];
            out[3 * r + 2] = pts[(size_t)idx * 3 + 2];
            out[3 * K + r] = (float)idx;   // indices exact: N = 2^23
        }
    }
}

extern "C" void kernel_launch(void* const* d_in, const int* in_sizes, int n_in,
                              void* d_out, int out_size, void* d_ws, size_t ws_size,
                              hipStream_t stream) {
    const float* pts = (const float*)d_in[0];   // (1, N, 3) float32
    const float* P1  = (const float*)d_in[1];   // (3,) float32
    const int*   Kp  = (const int*)d_in[2];     // scalar K (=64)
    float* out = (float*)d_out;
    const int N = in_sizes[0] / 3;

    uint32_t* hist = (uint32_t*)d_ws;                 // 8192 u32
    uint32_t* ctrl = hist + NB;                       // [0]=T, [1]=candCount (+pad)
    uint64_t* cand = (uint64_t*)(ctrl + 8);           // byte off 32800, 8B aligned

    knn_zero_kernel<<<(NB + 8 + 255) / 256, 256, 0, stream>>>(hist, NB + 8);
    knn_hist_kernel<<<GRID1, BLK, 0, stream>>>(pts, P1, hist, N);
    knn_thresh_kernel<<<1, 256, 0, stream>>>(hist, ctrl, Kp);
    knn_collect_kernel<<<GRID1, BLK, 0, stream>>>(pts, P1, ctrl, &ctrl[1], cand, N);
    knn_final_kernel<<<1, 256, 0, stream>>>(pts, ctrl, cand, Kp, out);
}